// PPO_43173011260132
// MI455X (gfx1250) — compile-verified
//
#include <hip/hip_runtime.h>
#include <hip/hip_bf16.h>

// PPO loss for MI455X (gfx1250). Layer-1 GEMM (K=4) mapped onto
// V_WMMA_F32_16X16X4_F32 (exact fp32 semantics); rank-1/2 heads fused in VALU;
// GAE = affine-map parallel scan (one wave per row); reductions via f32 atomics.
// Round 4: replace 32-wave serial GAE with 1024-wave shuffle-based suffix scan.

#define GAMMA    0.98f
#define LMBDA    0.95f
#define EPS_CLIP 0.1f
#define ADV_EPS  1e-8f

#define BSZ   1024
#define TLEN  256
#define DIN   4
#define HDIM  256
#define NTOT  (BSZ * TLEN)        // 262144
#define NTILES (NTOT / 16)        // 16384 (one wave each)

typedef float v2f __attribute__((ext_vector_type(2)));
typedef float v8f __attribute__((ext_vector_type(8)));

__device__ __forceinline__ v8f wmma4(v2f a, v2f b, v8f c) {
  // D(16x16) = A(16x4) * B(4x16) + C   -- exact fp32
  return __builtin_amdgcn_wmma_f32_16x16x4_f32(
      /*neg_a=*/false, a, /*neg_b=*/false, b,
      /*c_mod=*/(short)0, c, /*reuse_a=*/false, /*reuse_b=*/false);
}

// Value head for one 16-state tile of input x:
// returns (after butterfly) acc[r] = sum_n relu(hidden[r + hh*8, n] + b1[n]) * Wv[n]
// replicated across the 16 N-lanes of each half-wave.
__device__ __forceinline__ v8f head_value(const float* __restrict__ x,
                                          const float* __restrict__ W1,
                                          const float* __restrict__ b1,
                                          const float* __restrict__ Wv,
                                          int tile, int lq, int klo) {
  const int m0 = tile * 16 + lq;     // A-matrix row M = lq
  v2f A;
  A.x = x[m0 * DIN + klo];
  A.y = x[m0 * DIN + klo + 1];

  v8f acc = {};
  #pragma unroll 4
  for (int c = 0; c < 16; ++c) {     // 16 hidden-unit chunks of 16
    const int n = c * 16 + lq;       // B-matrix col N = lq
    v2f Bw;
    Bw.x = W1[klo * HDIM + n];
    Bw.y = W1[(klo + 1) * HDIM + n];
    v8f h = {};
    h = wmma4(A, Bw, h);
    const float bb = b1[n];
    const float wv = Wv[n];
    #pragma unroll
    for (int r = 0; r < 8; ++r)      // C layout: vgpr r -> state row r (+8 upper half)
      acc[r] += fmaxf(h[r] + bb, 0.f) * wv;
  }
  // butterfly over the 16 N-lanes (masks 1..8 keep half-waves separate)
  #pragma unroll
  for (int mk = 1; mk <= 8; mk <<= 1) {
    #pragma unroll
    for (int r = 0; r < 8; ++r)
      acc[r] += __shfl_xor(acc[r], mk, 32);
  }
  return acc;
}

// ---------------------------------------------------------------- K0: zero acc
__global__ void ppo_zero(float* __restrict__ acc) {
  if (threadIdx.x < 4) acc[threadIdx.x] = 0.f;
}

// ------------------------------------------- K1: values(s), values(s'), td, delta
__global__ void ppo_value_td(const float* __restrict__ s,
                             const float* __restrict__ sp,
                             const float* __restrict__ rew,
                             const float* __restrict__ dm,
                             const float* __restrict__ W1,
                             const float* __restrict__ b1,
                             const float* __restrict__ Wv,
                             const float* __restrict__ bv,
                             float* __restrict__ values,
                             float* __restrict__ td,
                             float* __restrict__ delta) {
  const int lane = threadIdx.x & 31;
  const int wave = threadIdx.x >> 5;
  const int tile = blockIdx.x * (blockDim.x >> 5) + wave;
  const int hh   = lane >> 4;        // half-wave: 0 or 1
  const int lq   = lane & 15;
  const int klo  = hh * 2;           // this lane's K pair: {klo, klo+1}

  // Two sequential passes -> only one accumulator set live at a time.
  const v8f accv  = head_value(s,  W1, b1, Wv, tile, lq, klo);
  const v8f accvp = head_value(sp, W1, b1, Wv, tile, lq, klo);

  if (lq == 0) {
    const float bvv   = bv[0];
    const int   mbase = tile * 16 + hh * 8;
    #pragma unroll
    for (int r = 0; r < 8; ++r) {
      const int   m   = mbase + r;
      const float v   = accv[r]  + bvv;
      const float vp  = accvp[r] + bvv;
      const float tdv = rew[m] + GAMMA * vp * dm[m];
      values[m] = v;
      td[m]     = tdv;
      delta[m]  = tdv - v;
    }
  }
}

// -------------------- K2: GAE as a wave-parallel suffix scan of affine maps
// g[t] = d[t] + c[t]*g[t+1]  ==  affine map (S,P): g -> S + P*g per segment;
// composition (left o right) = (S_l + P_l*S_r, P_l*P_r).
// One wave per batch row; lane l owns t in [8l, 8l+8).
__global__ void ppo_gae(const float* __restrict__ delta,
                        const float* __restrict__ dm,
                        float* __restrict__ adv,
                        float* __restrict__ acc) {
  const int lane = threadIdx.x & 31;
  const int wave = threadIdx.x >> 5;
  const int b    = blockIdx.x * (blockDim.x >> 5) + wave;
  const int base = b * TLEN + lane * 8;
  const float gl = GAMMA * LMBDA;

  // coalesced vector loads: 8 contiguous floats per lane
  const float4 d0 = *(const float4*)(delta + base);
  const float4 d1 = *(const float4*)(delta + base + 4);
  const float4 m0 = *(const float4*)(dm + base);
  const float4 m1 = *(const float4*)(dm + base + 4);
  float d[8] = {d0.x, d0.y, d0.z, d0.w, d1.x, d1.y, d1.z, d1.w};
  float c[8] = {gl * m0.x, gl * m0.y, gl * m0.z, gl * m0.w,
                gl * m1.x, gl * m1.y, gl * m1.z, gl * m1.w};

  // local segment affine map (right-to-left)
  float S = 0.f, P = 1.f;
  #pragma unroll
  for (int k = 7; k >= 0; --k) {
    S = d[k] + c[k] * S;
    P = P * c[k];
  }

  // inclusive suffix scan over lanes (Hillis-Steele toward higher lanes)
  float Si = S, Pi = P;
  #pragma unroll
  for (int off = 1; off < 32; off <<= 1) {
    const float Sn = __shfl_down(Si, off, 32);
    const float Pn = __shfl_down(Pi, off, 32);
    const bool ok = (lane + off) < 32;
    Si = ok ? (Si + Pi * Sn) : Si;
    Pi = ok ? (Pi * Pn) : Pi;
  }
  // exclusive: g entering this segment from the right = inclusive of lane+1
  float gright = __shfl_down(Si, 1, 32);
  if (lane == 31) gright = 0.f;

  // re-derive the 8 per-element advantages with the true right boundary
  float g = gright, lsum = 0.f, lsq = 0.f;
  float out[8];
  #pragma unroll
  for (int k = 7; k >= 0; --k) {
    g = d[k] + c[k] * g;
    out[k] = g;
    lsum += g;
    lsq  += g * g;
  }
  *(float4*)(adv + base)     = make_float4(out[0], out[1], out[2], out[3]);
  *(float4*)(adv + base + 4) = make_float4(out[4], out[5], out[6], out[7]);

  // wave-level reduction, one atomic pair per wave
  #pragma unroll
  for (int off = 16; off >= 1; off >>= 1) {
    lsum += __shfl_xor(lsum, off, 32);
    lsq  += __shfl_xor(lsq,  off, 32);
  }
  if (lane == 0) {
    atomicAdd(&acc[0], lsum);
    atomicAdd(&acc[1], lsq);
  }
}

// ----------------------------------- K3: policy forward + clipped surrogate + huber
__global__ void ppo_policy_loss(const float* __restrict__ s,
                                const int* __restrict__ act,
                                const float* __restrict__ prob_a,
                                const float* __restrict__ W1,
                                const float* __restrict__ b1,
                                const float* __restrict__ Wpi,
                                const float* __restrict__ bpi,
                                const float* __restrict__ values,
                                const float* __restrict__ td,
                                const float* __restrict__ adv,
                                float* __restrict__ acc) {
  // advantage normalization stats (global mean, unbiased std)
  const float ssum = acc[0];
  const float ssq  = acc[1];
  const float mean = ssum / (float)NTOT;
  const float var  = (ssq - ssum * mean) / (float)(NTOT - 1);
  const float stdv = sqrtf(fmaxf(var, 0.f));

  const int lane = threadIdx.x & 31;
  const int wave = threadIdx.x >> 5;
  const int tile = blockIdx.x * (blockDim.x >> 5) + wave;
  const int hh   = lane >> 4;
  const int lq   = lane & 15;
  const int klo  = hh * 2;
  const int m0   = tile * 16 + lq;

  v2f As;
  As.x = s[m0 * DIN + klo];
  As.y = s[m0 * DIN + klo + 1];

  v8f acc0 = {};   // logit 0 partials
  v8f acc1 = {};   // logit 1 partials

  #pragma unroll 4
  for (int c = 0; c < 16; ++c) {
    const int n = c * 16 + lq;
    v2f Bw;
    Bw.x = W1[klo * HDIM + n];
    Bw.y = W1[(klo + 1) * HDIM + n];
    v8f h = {};
    h = wmma4(As, Bw, h);
    const float bb = b1[n];
    const float w0 = Wpi[n * 2 + 0];
    const float w1 = Wpi[n * 2 + 1];
    #pragma unroll
    for (int r = 0; r < 8; ++r) {
      const float hv = fmaxf(h[r] + bb, 0.f);
      acc0[r] += hv * w0;
      acc1[r] += hv * w1;
    }
  }
  #pragma unroll
  for (int mk = 1; mk <= 8; mk <<= 1) {
    #pragma unroll
    for (int r = 0; r < 8; ++r) {
      acc0[r] += __shfl_xor(acc0[r], mk, 32);
      acc1[r] += __shfl_xor(acc1[r], mk, 32);
    }
  }

  float lsum = 0.f;
  if (lq == 0) {
    const float bpi0 = bpi[0], bpi1 = bpi[1];
    const int   mbase = tile * 16 + hh * 8;
    #pragma unroll
    for (int r = 0; r < 8; ++r) {
      const int   m  = mbase + r;
      const float l0 = acc0[r] + bpi0;
      const float l1 = acc1[r] + bpi1;
      const float mx = fmaxf(l0, l1);
      const float e0 = __expf(l0 - mx);
      const float e1 = __expf(l1 - mx);
      const float pia   = ((act[m] == 0) ? e0 : e1) / (e0 + e1);
      const float ratio = pia / prob_a[m];
      const float advn  = (adv[m] - mean) / (stdv + ADV_EPS);
      const float s1 = ratio * advn;
      const float rc = fminf(fmaxf(ratio, 1.f - EPS_CLIP), 1.f + EPS_CLIP);
      const float s2 = rc * advn;
      const float dv = values[m] - td[m];
      const float ad = fabsf(dv);
      const float hub = (ad < 1.f) ? (0.5f * dv * dv) : (ad - 0.5f);
      lsum += -fminf(s1, s2) + hub;
    }
  }
  lsum += __shfl_xor(lsum, 16, 32);   // combine lane 0 + lane 16
  if (lane == 0) atomicAdd(&acc[2], lsum);
}

// ---------------------------------------------------------------- K4: finalize
__global__ void ppo_finalize(const float* __restrict__ acc, float* __restrict__ out) {
  if (threadIdx.x == 0) out[0] = acc[2] / (float)NTOT;
}

extern "C" void kernel_launch(void* const* d_in, const int* in_sizes, int n_in,
                              void* d_out, int out_size, void* d_ws, size_t ws_size,
                              hipStream_t stream) {
  const float* s   = (const float*)d_in[0];
  const int*   act = (const int*)d_in[1];
  const float* rew = (const float*)d_in[2];
  const float* sp  = (const float*)d_in[3];
  const float* dm  = (const float*)d_in[4];
  const float* pa  = (const float*)d_in[5];
  const float* W1  = (const float*)d_in[6];
  const float* b1  = (const float*)d_in[7];
  const float* Wpi = (const float*)d_in[8];
  const float* bpi = (const float*)d_in[9];
  const float* Wv  = (const float*)d_in[10];
  const float* bv  = (const float*)d_in[11];

  float* values = (float*)d_ws;        // N
  float* td     = values + NTOT;       // N
  float* delta  = td + NTOT;           // N
  float* adv    = delta + NTOT;        // N
  float* acc    = adv + NTOT;          // 4 floats: {sum_adv, sumsq_adv, loss_sum, pad}

  ppo_zero<<<1, 32, 0, stream>>>(acc);
  ppo_value_td<<<NTILES / 8, 256, 0, stream>>>(s, sp, rew, dm, W1, b1, Wv, bv,
                                               values, td, delta);
  // one wave per batch row: 1024 waves, 8 waves per 256-thread block
  ppo_gae<<<BSZ / 8, 256, 0, stream>>>(delta, dm, adv, acc);
  ppo_policy_loss<<<NTILES / 8, 256, 0, stream>>>(s, act, pa, W1, b1, Wpi, bpi,
                                                  values, td, adv, acc);
  ppo_finalize<<<1, 32, 0, stream>>>(acc, (float*)d_out);
}